// LayerFlax_6975026889311
// MI455X (gfx1250) — compile-verified
//
#include <hip/hip_runtime.h>
#include <math.h>

// ---------------------------------------------------------------------------
// Problem constants (match reference)
// ---------------------------------------------------------------------------
#define NN     8192      // nodes
#define EE     131072    // edges
#define C0     64
#define C1     32
#define AA     4
#define RR     8
#define NQ     8
#define NL     3
#define NS     160       // 2*C0 + C1
#define NVC    128       // C0 + 2*C1
#define MIXW   288       // NS + NVC
#define AVG_NN 16.0f

typedef __attribute__((ext_vector_type(2))) float    v2f;
typedef __attribute__((ext_vector_type(8))) float    v8f;
typedef __attribute__((ext_vector_type(4))) uint32_t u32x4;
typedef __attribute__((ext_vector_type(8))) uint32_t u32x8;

// f32 WMMA: D(16x16) = A(16x4) * B(4x16) + C   (wave32, CDNA5)
__device__ inline v8f wmma4(v2f a, v2f b, v8f c) {
  return __builtin_amdgcn_wmma_f32_16x16x4_f32(false, a, false, b, (short)0, c,
                                               false, false);
}

__device__ inline float silu_f(float x) { return x / (1.0f + __expf(-x)); }

// low 32 bits of a generic LDS pointer == LDS byte offset (ISA 10.2 aperture)
__device__ inline uint32_t lds_offset(const void* p) {
  return (uint32_t)(uintptr_t)p;
}

// ---------------------------------------------------------------------------
// Generic strided f32 WMMA GEMM:  C[m,n] = scale * sum_k A[m,k] * B[k,n]
//   A element (m,k) at  A[m*lda + k*as]   (A pre-offset by blockIdx.y)
//   B element (k,n) at  B[k*ldb + n]
//   C element (m,n) at  C[m*ldc + n*cs]   (C pre-offset by blockIdx.y)
// one wave computes one 16x16 output tile; 8 waves / block
// ---------------------------------------------------------------------------
__global__ void gemm_wmma_f32(const float* __restrict__ A,
                              const float* __restrict__ B,
                              float* __restrict__ C,
                              int numTiles, int n16, int k4,
                              int lda, int as, int ldb, int ldc, int cs,
                              float scale) {
  const int lane = threadIdx.x & 31;
  const int wave = threadIdx.x >> 5;
  const int half = lane >> 4;
  const int l16  = lane & 15;
  int tile = blockIdx.x * 8 + wave;
  if (tile >= numTiles) return;
  const int mt = tile / n16;
  const int nt = tile % n16;

  A += (size_t)blockIdx.y;
  C += (size_t)blockIdx.y;
  const float* __restrict__ Arow = A + (size_t)(mt * 16 + l16) * lda;
  const int kb = 2 * half;

  v8f acc = {};
  for (int ks = 0; ks < k4; ++ks) {
    const int k0 = ks * 4 + kb;
    v2f a, b;
    a.x = Arow[(size_t)k0 * as];
    a.y = Arow[(size_t)(k0 + 1) * as];
    b.x = B[(size_t)k0 * ldb + nt * 16 + l16];
    b.y = B[(size_t)(k0 + 1) * ldb + nt * 16 + l16];
    acc = wmma4(a, b, acc);
  }
#pragma unroll
  for (int r = 0; r < 8; ++r) {
    const int row = mt * 16 + r + 8 * half;
    const int col = nt * 16 + l16;
    C[(size_t)row * ldc + (size_t)col * cs] = acc[r] * scale;
  }
}

// ---------------------------------------------------------------------------
// Species-gated skip connections (SIMT; A=4 species, per-node matrices)
// ---------------------------------------------------------------------------
__global__ void skip_s_kernel(const float* __restrict__ s_in,
                              const float* __restrict__ Wskip0,
                              const int* __restrict__ species,
                              float* __restrict__ skip_s) {
  const int idx = blockIdx.x * blockDim.x + threadIdx.x;  // N*C0
  const int n = idx >> 6, d = idx & 63;
  const float* __restrict__ W = Wskip0 + (size_t)species[n] * C0 * C0 + d;
  const float* __restrict__ srow = s_in + (size_t)n * C0;
  float acc = 0.f;
#pragma unroll 8
  for (int c = 0; c < C0; ++c) acc += srow[c] * W[c * C0];
  skip_s[idx] = acc;
}

__global__ void skip_v_kernel(const float* __restrict__ v_in,
                              const float* __restrict__ Wskip1,
                              const int* __restrict__ species,
                              float* __restrict__ skip_v) {
  const int idx = blockIdx.x * blockDim.x + threadIdx.x;  // N*C1
  const int n = idx >> 5, d = idx & 31;
  const float* __restrict__ W = Wskip1 + (size_t)species[n] * C1 * C1 + d;
  float ax = 0.f, ay = 0.f, az = 0.f;
#pragma unroll 8
  for (int c = 0; c < C1; ++c) {
    const float w = W[c * C1];
    const size_t b = ((size_t)n * C1 + c) * 3;
    ax += v_in[b + 0] * w;
    ay += v_in[b + 1] * w;
    az += v_in[b + 2] * w;
  }
  const size_t o = ((size_t)n * C1 + d) * 3;
  skip_v[o + 0] = ax;
  skip_v[o + 1] = ay;
  skip_v[o + 2] = az;
}

// ---------------------------------------------------------------------------
// Edge radial MLP (the dominant GEMM): mix = (silu(silu(radial@Wm1)@Wm2))@Wm3
// 128 edges / block, 8 waves.
//   Wm2 (64x64) -> LDS via one TDM tensor_load_to_lds (D# with LDS row pad:
//                  pad_interval = 64 dwords, pad_amount = 4 dwords -> ld 68)
//   Wm1 (8x64)  -> LDS via global_load_async_to_lds_b128 (ASYNCcnt)
//   Wm3 (64x288, L2-resident) stays global, warmed with global_prefetch_b8
// ---------------------------------------------------------------------------
__global__ void edge_mlp_kernel(const float* __restrict__ radial,
                                const float* __restrict__ Wm1,
                                const float* __restrict__ Wm2,
                                const float* __restrict__ Wm3,
                                float* __restrict__ mix) {
  __shared__ float sh_h[128][68];   // hidden activations (64 cols + pad)
  __shared__ float sh_w2[64 * 68];  // Wm2, row stride 68 (TDM-padded)
  __shared__ float sh_w1[8 * 68];   // Wm1, row stride 68

  const int lane = threadIdx.x & 31;
  const int wave = threadIdx.x >> 5;
  const int half = lane >> 4;
  const int l16  = lane & 15;
  const int kb   = 2 * half;
  const int e0   = blockIdx.x * 128;
  const int mrow = wave * 16;  // this wave's 16-row slab in LDS

  // ---- TDM: DMA Wm2 -> LDS with padded rows (TENSORcnt) --------------------
  if (wave == 0) {
    const uint64_t ga = (uint64_t)(uintptr_t)Wm2;
    u32x4 g0;
    g0[0] = 1u;                                            // count=1
    g0[1] = lds_offset(sh_w2);                             // lds_addr (bytes)
    g0[2] = (uint32_t)ga;                                  // global_addr[31:0]
    g0[3] = (uint32_t)((ga >> 32) & 0x1FFFFFFu) | (2u << 30);  // addr[56:32]|type=2
    u32x8 g1;
    g1[0] = (2u << 16)        // data_size = 4B
          | (1u << 20)        // pad_enable
          | (5u << 22)        // pad_interval: 64 dwords
          | (3u << 25);       // pad_amount:   4 dwords
    g1[1] = 4096u << 16;      // tensor_dim0 = 4096 (low 16)
    g1[2] = 1u << 16;         // tensor_dim0 hi=0 | tensor_dim1 = 1
    g1[3] = 4096u << 16;      // tensor_dim1 hi=0 | tile_dim0 = 4096
    g1[4] = 1u;               // tile_dim1 = 1, tile_dim2 = 0
    g1[5] = 4096u;            // tensor_dim0_stride low32
    g1[6] = 0u;
    g1[7] = 0u;
    asm volatile("tensor_load_to_lds %0, %1" ::"s"(g0), "s"(g1) : "memory");
  }

  // ---- async: Wm1 -> LDS (ASYNCcnt), one b128 per participating lane ------
  if (threadIdx.x < 128) {
    const int rrow = threadIdx.x >> 4;     // 8 rows
    const int blk  = threadIdx.x & 15;     // 16 x b128 per row
    const uint32_t loff = lds_offset(sh_w1) + (uint32_t)(rrow * 68 + blk * 4) * 4u;
    const float* gp = Wm1 + (rrow * C0 + blk * 4);
    asm volatile("global_load_async_to_lds_b128 %0, %1, off" ::"v"(loff),
                 "v"(gp)
                 : "memory");
  }

  // warm L2/L0 for the big streamed B panel (Wm3: 64x288)
  __builtin_prefetch(Wm3 + (size_t)threadIdx.x * 72, 0, 0);

  // ---- layer 1: [16x8] @ [8x64], silu -> LDS -------------------------------
  const float* __restrict__ Ar = radial + (size_t)(e0 + mrow + l16) * RR;
  asm volatile("s_wait_asynccnt 0x0" ::: "memory");
  if (wave == 0) __builtin_amdgcn_s_wait_tensorcnt(0);
  __syncthreads();  // all waves see Wm1/Wm2 in LDS

#pragma unroll
  for (int nt = 0; nt < 4; ++nt) {
    v8f acc = {};
#pragma unroll
    for (int ks = 0; ks < 2; ++ks) {
      const int k0 = ks * 4 + kb;
      v2f a, b;
      a.x = Ar[k0];
      a.y = Ar[k0 + 1];
      b.x = sh_w1[k0 * 68 + nt * 16 + l16];
      b.y = sh_w1[(k0 + 1) * 68 + nt * 16 + l16];
      acc = wmma4(a, b, acc);
    }
#pragma unroll
    for (int r = 0; r < 8; ++r)
      sh_h[mrow + r + 8 * half][nt * 16 + l16] = silu_f(acc[r]);
  }
  __syncthreads();

  // ---- layer 2: [16x64] @ [64x64], silu -> LDS -----------------------------
  v8f acc2[4];
#pragma unroll
  for (int nt = 0; nt < 4; ++nt) {
    v8f acc = {};
    for (int ks = 0; ks < 16; ++ks) {
      const int k0 = ks * 4 + kb;
      v2f a, b;
      a.x = sh_h[mrow + l16][k0];
      a.y = sh_h[mrow + l16][k0 + 1];
      b.x = sh_w2[k0 * 68 + nt * 16 + l16];
      b.y = sh_w2[(k0 + 1) * 68 + nt * 16 + l16];
      acc = wmma4(a, b, acc);
    }
    acc2[nt] = acc;
  }
  __syncthreads();
#pragma unroll
  for (int nt = 0; nt < 4; ++nt)
#pragma unroll
    for (int r = 0; r < 8; ++r)
      sh_h[mrow + r + 8 * half][nt * 16 + l16] = silu_f(acc2[nt][r]);
  __syncthreads();

  // ---- layer 3: [16x64] @ [64x288] -> mix ----------------------------------
  for (int nt = 0; nt < 18; ++nt) {
    v8f acc = {};
    for (int ks = 0; ks < 16; ++ks) {
      const int k0 = ks * 4 + kb;
      v2f a, b;
      a.x = sh_h[mrow + l16][k0];
      a.y = sh_h[mrow + l16][k0 + 1];
      b.x = Wm3[k0 * MIXW + nt * 16 + l16];
      b.y = Wm3[(k0 + 1) * MIXW + nt * 16 + l16];
      acc = wmma4(a, b, acc);
    }
#pragma unroll
    for (int r = 0; r < 8; ++r) {
      const int e = e0 + mrow + r + 8 * half;
      mix[(size_t)e * MIXW + nt * 16 + l16] = acc[r];
    }
  }
}

// ---------------------------------------------------------------------------
// Edge message build + mix-scale + scatter-add (fused; fp32 global atomics)
// ---------------------------------------------------------------------------
__global__ void edge_apply_kernel(const float* __restrict__ s1,
                                  const float* __restrict__ v1,
                                  const float* __restrict__ Y0,
                                  const float* __restrict__ Y1,
                                  const int* __restrict__ senders,
                                  const int* __restrict__ receivers,
                                  const float* __restrict__ mix,
                                  float* __restrict__ agg_s,
                                  float* __restrict__ agg_v) {
  const int e = blockIdx.x * blockDim.x + threadIdx.x;
  if (e >= EE) return;
  const int snd = senders[e];
  const int rcv = receivers[e];
  const float y0 = Y0[e];
  const float y1x = Y1[e * 3 + 0], y1y = Y1[e * 3 + 1], y1z = Y1[e * 3 + 2];
  const float* __restrict__ mx = mix + (size_t)e * MIXW;
  const float* __restrict__ ss = s1 + (size_t)snd * C0;
  const float* __restrict__ vv = v1 + (size_t)snd * C1 * 3;
  float* __restrict__ as_ = agg_s + (size_t)rcv * NS;
  float* __restrict__ av_ = agg_v + (size_t)rcv * NVC * 3;
  const float rs3 = 0.57735026918962576f;

#pragma unroll 4
  for (int c = 0; c < C0; ++c) {
    const float ms = ss[c];
    atomicAdd(&as_[c], ms * mx[c]);                    // msg_s: ms
    atomicAdd(&as_[C0 + c], ms * y0 * mx[C0 + c]);     // msg_s: ms*Y0
    const float g = mx[NS + C1 + c];                   // msg_v: sY1 (ch 32..95)
    atomicAdd(&av_[(C1 + c) * 3 + 0], ms * y1x * g);
    atomicAdd(&av_[(C1 + c) * 3 + 1], ms * y1y * g);
    atomicAdd(&av_[(C1 + c) * 3 + 2], ms * y1z * g);
  }
#pragma unroll 4
  for (int c = 0; c < C1; ++c) {
    const float mvx = vv[c * 3 + 0], mvy = vv[c * 3 + 1], mvz = vv[c * 3 + 2];
    const float dot = (mvx * y1x + mvy * y1y + mvz * y1z) * rs3;
    atomicAdd(&as_[2 * C0 + c], dot * mx[2 * C0 + c]); // msg_s: dot
    const float g0 = mx[NS + c];                       // msg_v: mv (ch 0..31)
    atomicAdd(&av_[c * 3 + 0], mvx * g0);
    atomicAdd(&av_[c * 3 + 1], mvy * g0);
    atomicAdd(&av_[c * 3 + 2], mvz * g0);
    const float g2 = mx[NS + C1 + C0 + c];             // msg_v: vY0 (ch 96..127)
    atomicAdd(&av_[(C1 + C0 + c) * 3 + 0], mvx * y0 * g2);
    atomicAdd(&av_[(C1 + C0 + c) * 3 + 1], mvy * y0 * g2);
    atomicAdd(&av_[(C1 + C0 + c) * 3 + 2], mvz * y0 * g2);
  }
}

// ---------------------------------------------------------------------------
// Node scalar path: s3 = (x_node (x) s2) @ Wt0 + skip_s  ->  out_s
//                   feats = s3 @ Wq  (Wq padded to 16 cols with zeros)
// A-fragments for the tensor product are synthesized on the fly.
// ---------------------------------------------------------------------------
__global__ void node_ts_kernel(const float* __restrict__ s2,
                               const float* __restrict__ x_node,
                               const float* __restrict__ Wt0,
                               const float* __restrict__ skip_s,
                               const float* __restrict__ Wq,
                               float* __restrict__ out_s,
                               float* __restrict__ feats) {
  __shared__ float sh[8][16][68];
  const int lane = threadIdx.x & 31;
  const int wave = threadIdx.x >> 5;
  const int half = lane >> 4;
  const int l16  = lane & 15;
  const int kb   = 2 * half;
  const int mt   = blockIdx.x * 8 + wave;
  const int row  = mt * 16 + l16;
  const float* __restrict__ xr = x_node + (size_t)row * AA;
  const float* __restrict__ sr = s2 + (size_t)row * C0;

#pragma unroll
  for (int nt = 0; nt < 4; ++nt) {
    v8f acc = {};
    for (int ks = 0; ks < 64; ++ks) {  // K = A*C0 = 256
      const int k0 = ks * 4 + kb;
      v2f a, b;
      a.x = xr[k0 >> 6] * sr[k0 & 63];
      a.y = xr[(k0 + 1) >> 6] * sr[(k0 + 1) & 63];
      b.x = Wt0[k0 * C0 + nt * 16 + l16];
      b.y = Wt0[(k0 + 1) * C0 + nt * 16 + l16];
      acc = wmma4(a, b, acc);
    }
#pragma unroll
    for (int r = 0; r < 8; ++r) {
      const int rr = mt * 16 + r + 8 * half;
      const int cc = nt * 16 + l16;
      const float val = acc[r] + skip_s[(size_t)rr * C0 + cc];
      out_s[(size_t)rr * C0 + cc] = val;
      sh[wave][r + 8 * half][cc] = val;
    }
  }
  __syncthreads();

  // feats = s3 @ Wq : [16x64] @ [64x8] (cols 8..15 zero-padded)
  v8f accq = {};
  for (int ks = 0; ks < 16; ++ks) {
    const int k0 = ks * 4 + kb;
    v2f a, b;
    a.x = sh[wave][l16][k0];
    a.y = sh[wave][l16][k0 + 1];
    b.x = (l16 < NQ) ? Wq[k0 * NQ + l16] : 0.0f;
    b.y = (l16 < NQ) ? Wq[(k0 + 1) * NQ + l16] : 0.0f;
    accq = wmma4(a, b, accq);
  }
  if (l16 < NQ) {
#pragma unroll
    for (int r = 0; r < 8; ++r)
      feats[(size_t)(mt * 16 + r + 8 * half) * NQ + l16] = accq[r];
  }
}

// ---------------------------------------------------------------------------
// Node vector path: v3 = (x_node (x) v2) @ Wt1 + skip_v -> out_v (per comp)
// ---------------------------------------------------------------------------
__global__ void node_tv_kernel(const float* __restrict__ v2,
                               const float* __restrict__ x_node,
                               const float* __restrict__ Wt1,
                               const float* __restrict__ skip_v,
                               float* __restrict__ out_v) {
  const int lane = threadIdx.x & 31;
  const int wave = threadIdx.x >> 5;
  const int half = lane >> 4;
  const int l16  = lane & 15;
  const int kb   = 2 * half;
  const int comp = blockIdx.y;
  const int mt   = blockIdx.x * 8 + wave;
  const int row  = mt * 16 + l16;
  const float* __restrict__ xr = x_node + (size_t)row * AA;
  const float* __restrict__ vr = v2 + (size_t)row * C1 * 3 + comp;

#pragma unroll
  for (int nt = 0; nt < 2; ++nt) {
    v8f acc = {};
    for (int ks = 0; ks < 32; ++ks) {  // K = A*C1 = 128
      const int k0 = ks * 4 + kb;
      v2f a, b;
      a.x = xr[k0 >> 5] * vr[(k0 & 31) * 3];
      a.y = xr[(k0 + 1) >> 5] * vr[((k0 + 1) & 31) * 3];
      b.x = Wt1[k0 * C1 + nt * 16 + l16];
      b.y = Wt1[(k0 + 1) * C1 + nt * 16 + l16];
      acc = wmma4(a, b, acc);
    }
#pragma unroll
    for (int r = 0; r < 8; ++r) {
      const int rr = mt * 16 + r + 8 * half;
      const int cc = nt * 16 + l16;
      const size_t o = ((size_t)rr * C1 + cc) * 3 + comp;
      out_v[o] = acc[r] + skip_v[o];
    }
  }
}

// ---------------------------------------------------------------------------
// 8-qubit circuit: one wave32 / node, 256 amplitudes = 8 floats/lane.
// flat amp index = lane*8 + j ; qubit q <-> bit (7-q) of index
//   qubits 0..4 -> lane bits 4..0 (cross-lane: __shfl_xor -> ds permute path)
//   qubits 5..7 -> register bits 2..0 (in-lane)
// ---------------------------------------------------------------------------
template <int Q>
__device__ inline void ry_gate(float (&amp)[8], int lane, float theta) {
  const float c = __cosf(0.5f * theta);
  const float s = __sinf(0.5f * theta);
  if constexpr (Q < 5) {
    constexpr int lm = 1 << (4 - Q);
    const int bit = (lane >> (4 - Q)) & 1;
    const float sg = bit ? s : -s;
#pragma unroll
    for (int j = 0; j < 8; ++j) {
      const float p = __shfl_xor(amp[j], lm, 32);
      amp[j] = fmaf(sg, p, c * amp[j]);
    }
  } else {
    constexpr int m = 1 << (7 - Q);
#pragma unroll
    for (int j = 0; j < 8; ++j) {
      if ((j & m) == 0) {
        const float a0 = amp[j], a1 = amp[j | m];
        amp[j]     = c * a0 - s * a1;
        amp[j | m] = s * a0 + c * a1;
      }
    }
  }
}

template <int C, int T>
__device__ inline void cnot_gate(float (&amp)[8], int lane) {
  constexpr int bc = 7 - C, bt = 7 - T;
  if constexpr (bc >= 3 && bt >= 3) {          // ctrl & tgt on lane bits
    constexpr int lmt = 1 << (bt - 3);
    const int ctrl = (lane >> (bc - 3)) & 1;
#pragma unroll
    for (int j = 0; j < 8; ++j) {
      const float p = __shfl_xor(amp[j], lmt, 32);
      if (ctrl) amp[j] = p;
    }
  } else if constexpr (bc >= 3 && bt < 3) {    // ctrl on lane, tgt in regs
    constexpr int m = 1 << bt;
    const int ctrl = (lane >> (bc - 3)) & 1;
#pragma unroll
    for (int j = 0; j < 8; ++j) {
      if ((j & m) == 0) {
        const float a0 = amp[j], a1 = amp[j | m];
        amp[j]     = ctrl ? a1 : a0;
        amp[j | m] = ctrl ? a0 : a1;
      }
    }
  } else if constexpr (bc < 3 && bt >= 3) {    // ctrl in regs, tgt on lane
    constexpr int lmt = 1 << (bt - 3);
#pragma unroll
    for (int j = 0; j < 8; ++j) {
      const float p = __shfl_xor(amp[j], lmt, 32);
      if (j & (1 << bc)) amp[j] = p;
    }
  } else {                                     // both in regs
    constexpr int mc = 1 << bc, mt2 = 1 << bt;
#pragma unroll
    for (int j = 0; j < 8; ++j) {
      if ((j & mc) && !(j & mt2)) {
        const float t = amp[j];
        amp[j]       = amp[j | mt2];
        amp[j | mt2] = t;
      }
    }
  }
}

template <int Q = 0>
__device__ inline void ry_all(float (&amp)[8], int lane, const float (&th)[8]) {
  ry_gate<Q>(amp, lane, th[Q]);
  if constexpr (Q < 7) ry_all<Q + 1>(amp, lane, th);
}
template <int Q = 0>
__device__ inline void cnot_ring(float (&amp)[8], int lane) {
  cnot_gate<Q, (Q + 1) & 7>(amp, lane);
  if constexpr (Q < 7) cnot_ring<Q + 1>(amp, lane);
}

__global__ void qcirc_kernel(const float* __restrict__ feats,
                             const float* __restrict__ wqml,
                             const float* __restrict__ Wo,
                             const float* __restrict__ bo,
                             float* __restrict__ out) {
  const int lane = threadIdx.x & 31;
  const int node = (blockIdx.x * blockDim.x + threadIdx.x) >> 5;
  if (node >= NN) return;

  float amp[8];
#pragma unroll
  for (int j = 0; j < 8; ++j) amp[j] = 0.0f;
  if (lane == 0) amp[0] = 1.0f;

  float th[8];
#pragma unroll
  for (int q = 0; q < NQ; ++q) th[q] = feats[(size_t)node * NQ + q];
  ry_all(amp, lane, th);

  for (int l = 0; l < NL; ++l) {
    float wl[8];
#pragma unroll
    for (int q = 0; q < NQ; ++q) wl[q] = wqml[l * NQ + q];
    ry_all(amp, lane, wl);
    cnot_ring(amp, lane);
  }

  // <Z_q> readout
  float z[8];
#pragma unroll
  for (int q = 0; q < 8; ++q) z[q] = 0.0f;
#pragma unroll
  for (int j = 0; j < 8; ++j) {
    const float p = amp[j] * amp[j];
#pragma unroll
    for (int q = 0; q < 8; ++q) {
      const int bit = (q < 5) ? ((lane >> (4 - q)) & 1) : ((j >> (7 - q)) & 1);
      z[q] += bit ? -p : p;
    }
  }
#pragma unroll
  for (int off = 16; off >= 1; off >>= 1)
#pragma unroll
    for (int q = 0; q < 8; ++q) z[q] += __shfl_xor(z[q], off, 32);

  float o = bo[0];
#pragma unroll
  for (int q = 0; q < 8; ++q) o += z[q] * Wo[q];
  if (lane == 0) out[node] = o;
}

// ---------------------------------------------------------------------------
// Orchestration
// ---------------------------------------------------------------------------
extern "C" void kernel_launch(void* const* d_in, const int* in_sizes, int n_in,
                              void* d_out, int out_size, void* d_ws,
                              size_t ws_size, hipStream_t stream) {
  const float* s_in    = (const float*)d_in[0];
  const float* v_in    = (const float*)d_in[1];
  const float* x_node  = (const float*)d_in[2];
  const float* radial  = (const float*)d_in[3];
  const float* Y0      = (const float*)d_in[4];
  const float* Y1      = (const float*)d_in[5];
  const int*   species = (const int*)d_in[6];
  const int*   senders = (const int*)d_in[7];
  const int*   recvs   = (const int*)d_in[8];
  const float* Wskip0  = (const float*)d_in[9];
  const float* Wskip1  = (const float*)d_in[10];
  const float* Wu0     = (const float*)d_in[11];
  const float* Wu1     = (const float*)d_in[12];
  const float* Wm1     = (const float*)d_in[13];
  const float* Wm2     = (const float*)d_in[14];
  const float* Wm3     = (const float*)d_in[15];
  const float* Wd0     = (const float*)d_in[16];
  const float* Wd1     = (const float*)d_in[17];
  const float* Wt0     = (const float*)d_in[18];
  const float* Wt1     = (const float*)d_in[19];
  const float* Wq      = (const float*)d_in[20];
  const float* wqml    = (const float*)d_in[21];
  const float* Wo      = (const float*)d_in[22];
  const float* bo      = (const float*)d_in[23];

  float* out0  = (float*)d_out;            // (N,1)
  float* out_s = out0 + NN;                // (N,C0)
  float* out_v = out_s + (size_t)NN * C0;  // (N,C1,3)

  float* w = (float*)d_ws;
  float* ws_s1     = w; w += (size_t)NN * C0;
  float* ws_v1     = w; w += (size_t)NN * C1 * 3;
  float* ws_skip_s = w; w += (size_t)NN * C0;
  float* ws_skip_v = w; w += (size_t)NN * C1 * 3;
  float* ws_agg_s  = w; w += (size_t)NN * NS;
  float* ws_agg_v  = w; w += (size_t)NN * NVC * 3;
  float* ws_s2     = w; w += (size_t)NN * C0;
  float* ws_v2     = w; w += (size_t)NN * C1 * 3;
  float* ws_feats  = w; w += (size_t)NN * NQ;
  float* ws_mix    = w; w += (size_t)EE * MIXW;

  hipMemsetAsync(ws_agg_s, 0, (size_t)NN * NS * sizeof(float), stream);
  hipMemsetAsync(ws_agg_v, 0, (size_t)NN * NVC * 3 * sizeof(float), stream);

  // s1 = s_in @ Wu0 : M=8192, N=64, K=64 -> 2048 tiles
  gemm_wmma_f32<<<dim3(256, 1), 256, 0, stream>>>(
      s_in, Wu0, ws_s1, 2048, 4, 16, C0, 1, C0, C0, 1, 1.0f);
  // v1 = v_in @ Wu1 per component : M=8192, N=32, K=32 -> 1024 tiles
  gemm_wmma_f32<<<dim3(128, 3), 256, 0, stream>>>(
      v_in, Wu1, ws_v1, 1024, 2, 8, C1 * 3, 3, C1, C1 * 3, 3, 1.0f);

  skip_s_kernel<<<NN * C0 / 256, 256, 0, stream>>>(s_in, Wskip0, species,
                                                   ws_skip_s);
  skip_v_kernel<<<NN * C1 / 256, 256, 0, stream>>>(v_in, Wskip1, species,
                                                   ws_skip_v);

  edge_mlp_kernel<<<EE / 128, 256, 0, stream>>>(radial, Wm1, Wm2, Wm3, ws_mix);
  edge_apply_kernel<<<EE / 256, 256, 0, stream>>>(ws_s1, ws_v1, Y0, Y1, senders,
                                                  recvs, ws_mix, ws_agg_s,
                                                  ws_agg_v);

  // s2 = agg_s @ Wd0 / 16 : K=160
  gemm_wmma_f32<<<dim3(256, 1), 256, 0, stream>>>(
      ws_agg_s, Wd0, ws_s2, 2048, 4, 40, NS, 1, C0, C0, 1, 1.0f / AVG_NN);
  // v2 = agg_v @ Wd1 / 16 per component : K=128
  gemm_wmma_f32<<<dim3(128, 3), 256, 0, stream>>>(
      ws_agg_v, Wd1, ws_v2, 1024, 2, 32, NVC * 3, 3, C1, C1 * 3, 3,
      1.0f / AVG_NN);

  node_ts_kernel<<<64, 256, 0, stream>>>(ws_s2, x_node, Wt0, ws_skip_s, Wq,
                                         out_s, ws_feats);
  node_tv_kernel<<<dim3(64, 3), 256, 0, stream>>>(ws_v2, x_node, Wt1,
                                                  ws_skip_v, out_v);

  qcirc_kernel<<<NN / 8, 256, 0, stream>>>(ws_feats, wqml, Wo, bo, out0);

  (void)in_sizes; (void)n_in; (void)out_size; (void)ws_size;
}